// AWBNet_wo_R2_36936718745841
// MI455X (gfx1250) — compile-verified
//
#include <hip/hip_runtime.h>

typedef __attribute__((ext_vector_type(2))) float v2f;
typedef __attribute__((ext_vector_type(8))) float v8f;

#define HIST_K 12288   // 3*64*64
#define HIDDEN 256
#define M_OUT  27
#define BATCH  16
#define KC     1024    // K-chunk staged in LDS: 16*1024*4B = 64 KB per WGP

// ---------------------------------------------------------------------------
// Kernel 0: zero the feat accumulator (harness poisons ws; we accumulate atomically)
// ---------------------------------------------------------------------------
__global__ void zero_feat_kernel(float* __restrict__ feat) {
    int i = blockIdx.x * blockDim.x + threadIdx.x;
    if (i < BATCH * HIDDEN) feat[i] = 0.0f;
}

// ---------------------------------------------------------------------------
// Kernel 1: feat[b][n] += sum_{k in chunk} hist[b][k] * W1[k][n]
// Grid: 12 workgroups (one 1024-wide K chunk each) x 512 threads (16 waves).
// Wave w owns N-tile [16w, 16w+16). M=16 = batch. V_WMMA_F32_16X16X4_F32.
// hist chunk is staged in LDS (coalesced float4), W1 streamed from global.
// Partial 16x16 tiles accumulated into feat with global_atomic_add_f32.
// ---------------------------------------------------------------------------
__global__ __launch_bounds__(512) void gemm1_wmma_kernel(
    const float* __restrict__ hist, const float* __restrict__ W1,
    float* __restrict__ feat)
{
    __shared__ float hs[BATCH * KC];

    const int tid   = threadIdx.x;
    const int lane  = tid & 31;
    const int wave  = tid >> 5;            // 0..15 -> N tile
    const int n0    = wave * 16;
    const int mrow  = lane & 15;           // M row (batch) / N col this lane covers
    const int khalf = (lane >> 4) * 2;     // lanes 16..31 hold K+2, K+3
    const int k0    = blockIdx.x * KC;     // this workgroup's K chunk

    // Stage hist[:, k0:k0+KC] into LDS; rows contiguous -> coalesced float4 loads
    for (int i = tid; i < BATCH * (KC / 4); i += 512) {
        const int r = i / (KC / 4);
        const int c = (i % (KC / 4)) * 4;
        *(float4*)(hs + r * KC + c) =
            *(const float4*)(hist + (size_t)r * HIST_K + k0 + c);
    }
    __syncthreads();

    v8f acc = {};
    for (int kk = 0; kk < KC; kk += 4) {
        const int ka = kk + khalf;
        // A: 16x4 fp32 tile of hist (from LDS) — two consecutive K per lane
        v2f a = *(const v2f*)(hs + mrow * KC + ka);
        // B: 4x16 fp32 tile of W1 (row-major [K][N]) — K rows striped over lane halves
        v2f b;
        b.x = W1[(size_t)(k0 + ka + 0) * HIDDEN + n0 + mrow];
        b.y = W1[(size_t)(k0 + ka + 1) * HIDDEN + n0 + mrow];
        acc = __builtin_amdgcn_wmma_f32_16x16x4_f32(
            /*neg_a=*/false, a, /*neg_b=*/false, b,
            /*c_mod=*/(short)0, acc, /*reuse_a=*/false, /*reuse_b=*/false);
    }

    // D layout: VGPR v -> row (v + 8*(lane>=16)), col = lane&15 within the tile
    const int col   = n0 + mrow;
    const int rbase = (lane >> 4) * 8;
#pragma unroll
    for (int v = 0; v < 8; ++v) {
        atomicAdd(&feat[(size_t)(rbase + v) * HIDDEN + col], acc[v]);
    }
}

// ---------------------------------------------------------------------------
// Kernel 2: m[b][j] = b2[j] + sum_k relu(feat[b][k] + b1[k]) * W2[k][j]
// (bias + ReLU of GEMM1 folded here so the atomic accumulation needs no extra pass)
// ---------------------------------------------------------------------------
__global__ void gemm2_kernel(const float* __restrict__ feat,
                             const float* __restrict__ b1,
                             const float* __restrict__ W2,
                             const float* __restrict__ b2,
                             float* __restrict__ m)
{
    const int t = blockIdx.x * blockDim.x + threadIdx.x;
    if (t >= BATCH * M_OUT) return;
    const int b = t / M_OUT;
    const int j = t % M_OUT;
    float s = b2[j];
    for (int k = 0; k < HIDDEN; ++k) {
        const float f = fmaxf(feat[b * HIDDEN + k] + b1[k], 0.0f);
        s += f * W2[k * M_OUT + j];
    }
    m[t] = s;
}

// ---------------------------------------------------------------------------
// Kernel 3: streaming per-pixel polynomial color mapping (the ~100 MB stage).
// 4 pixels per thread -> 3x float4 (b128) loads + 3x float4 stores, coalesced.
// m[27] is uniform per block -> scalar loads.
// ---------------------------------------------------------------------------
__global__ __launch_bounds__(256) void apply_map_kernel(
    const float* __restrict__ x, const float* __restrict__ m,
    float* __restrict__ y)
{
    const int b   = blockIdx.x >> 8;          // 256 blocks per batch image
    const int blk = blockIdx.x & 255;
    const size_t pix0 = (size_t)b * (512 * 512) + (size_t)blk * 1024 +
                        (size_t)threadIdx.x * 4;

    float mm[27];
#pragma unroll
    for (int i = 0; i < 27; ++i) mm[i] = m[b * M_OUT + i];

    const float* xp = x + pix0 * 3;
    float*       yp = y + pix0 * 3;

    float in[12];
    *(float4*)&in[0] = *(const float4*)(xp + 0);
    *(float4*)&in[4] = *(const float4*)(xp + 4);
    *(float4*)&in[8] = *(const float4*)(xp + 8);

    float out[12];
#pragma unroll
    for (int p = 0; p < 4; ++p) {
        const float r  = in[p * 3 + 0];
        const float g  = in[p * 3 + 1];
        const float bb = in[p * 3 + 2];
#pragma unroll
        for (int c = 0; c < 3; ++c) {
            out[p * 3 + c] =
                r * mm[0 + c] + g * mm[3 + c] + bb * mm[6 + c] +
                r * r * mm[9 + c] + g * g * mm[12 + c] + bb * bb * mm[15 + c] +
                r * g * mm[18 + c] + r * bb * mm[21 + c] + g * bb * mm[24 + c];
        }
    }

    *(float4*)(yp + 0) = *(const float4*)&out[0];
    *(float4*)(yp + 4) = *(const float4*)&out[4];
    *(float4*)(yp + 8) = *(const float4*)&out[8];
}

// ---------------------------------------------------------------------------
extern "C" void kernel_launch(void* const* d_in, const int* in_sizes, int n_in,
                              void* d_out, int out_size, void* d_ws, size_t ws_size,
                              hipStream_t stream) {
    const float* x    = (const float*)d_in[0];  // [16,512,512,3]
    const float* hist = (const float*)d_in[1];  // [16,3,64,64] -> [16,12288]
    const float* W1   = (const float*)d_in[2];  // [12288,256]
    const float* b1   = (const float*)d_in[3];  // [256]
    const float* W2   = (const float*)d_in[4];  // [256,27]
    const float* b2   = (const float*)d_in[5];  // [27]

    float* feat = (float*)d_ws;                 // 16*256 floats (pre-activation sums)
    float* m    = feat + BATCH * HIDDEN;        // 16*27 floats

    zero_feat_kernel<<<(BATCH * HIDDEN + 255) / 256, 256, 0, stream>>>(feat);
    gemm1_wmma_kernel<<<HIST_K / KC, 512, 0, stream>>>(hist, W1, feat);
    gemm2_kernel<<<(BATCH * M_OUT + 255) / 256, 256, 0, stream>>>(feat, b1, W2, b2, m);
    apply_map_kernel<<<16 * 256, 256, 0, stream>>>(x, m, (float*)d_out);
}